// FCOSPostProcessor_48747878809966
// MI455X (gfx1250) — compile-verified
//
#include <hip/hip_runtime.h>
#include <stdint.h>

#define NB 2
#define T_TOT 856
#define POST_N 100
#define PRE_T 0.05f
#define NMS_T 0.5f
#define SORT_N 1024

typedef float v2f __attribute__((ext_vector_type(2)));
typedef float v8f __attribute__((ext_vector_type(8)));

// ---- 64-bit key: descending score, then ascending index (JAX stable tie-break) ----
__device__ __forceinline__ unsigned long long packKey(float f, unsigned idx) {
  unsigned u = __float_as_uint(f);
  u = (u & 0x80000000u) ? ~u : (u | 0x80000000u);
  return ((unsigned long long)u << 32) | (unsigned long long)(0xFFFFFFFFu - idx);
}
__device__ __forceinline__ unsigned keyIdx(unsigned long long k) {
  return 0xFFFFFFFFu - (unsigned)(k & 0xFFFFFFFFull);
}

__device__ __forceinline__ float quadAreaF(const float* qx, const float* qy) {
  float s = 0.f;
#pragma unroll
  for (int k = 0; k < 4; ++k) { int n = (k + 1) & 3; s += qx[k]*qy[n] - qx[n]*qy[k]; }
  return 0.5f * fabsf(s);
}

// Sutherland-Hodgman clip of quad A by quad B, faithful to the JAX reference
// (index clamping, drop-mode writes, 1e-9 denominator guard). All local arrays
// use static indices / unrolled select chains so everything stays in VGPRs.
__device__ float polyInterArea(const float* Ax, const float* Ay,
                               const float* Bx, const float* By) {
  float sB = 0.f;
#pragma unroll
  for (int e = 0; e < 4; ++e) { int n = (e + 1) & 3; sB += Bx[e]*By[n] - Bx[n]*By[e]; }
  float orient = (sB >= 0.f) ? 1.f : -1.f;

  float Vx[8], Vy[8];
#pragma unroll
  for (int k = 0; k < 8; ++k) { Vx[k] = (k < 4) ? Ax[k] : 0.f; Vy[k] = (k < 4) ? Ay[k] : 0.f; }
  int n = 4;

#pragma unroll
  for (int e = 0; e < 4; ++e) {
    float p1x = Bx[e], p1y = By[e];
    int e2 = (e + 1) & 3;
    float dx = Bx[e2] - p1x, dy = By[e2] - p1y;
    float Ox[8], Oy[8];
#pragma unroll
    for (int s = 0; s < 8; ++s) { Ox[s] = 0.f; Oy[s] = 0.f; }
    int cnt = 0;
#pragma unroll
    for (int k = 0; k < 8; ++k) {
      float cx = Vx[k], cy = Vy[k];
      int pidx = (k == 0) ? (n - 1) : (k - 1);
      pidx = pidx < 0 ? 0 : (pidx > 7 ? 7 : pidx);
      float px = Vx[0], py = Vy[0];
#pragma unroll
      for (int s = 0; s < 8; ++s) { if (pidx == s) { px = Vx[s]; py = Vy[s]; } }
      float scv = orient * (dx*(cy - p1y) - dy*(cx - p1x));
      float spv = orient * (dx*(py - p1y) - dy*(px - p1x));
      bool cin = (scv >= 0.f), pin = (spv >= 0.f);
      float denom = spv - scv;
      float t = spv / ((fabsf(denom) > 1e-9f) ? denom : 1e-9f);
      float ipx = px + t*(cx - px), ipy = py + t*(cy - py);
      bool valid = (k < n);
      bool emitI = valid && (cin != pin);
#pragma unroll
      for (int s = 0; s < 8; ++s) { if (emitI && cnt == s) { Ox[s] = ipx; Oy[s] = ipy; } }
      cnt += emitI ? 1 : 0;
      bool emitC = valid && cin;
#pragma unroll
      for (int s = 0; s < 8; ++s) { if (emitC && cnt == s) { Ox[s] = cx; Oy[s] = cy; } }
      cnt += emitC ? 1 : 0;
    }
#pragma unroll
    for (int s = 0; s < 8; ++s) { Vx[s] = Ox[s]; Vy[s] = Oy[s]; }
    n = cnt;
  }
  float a = 0.f;
#pragma unroll
  for (int k = 0; k < 8; ++k) {
    int kn = (k < 7) ? (k + 1) : 7;      // JAX gather clamp
    bool last = (k == n - 1);
    float nx = last ? Vx[0] : Vx[kn];
    float ny = last ? Vy[0] : Vy[kn];
    if (k < n) a += Vx[k]*ny - nx*Vy[k];
  }
  return 0.5f * fabsf(a);
}

// ---- Stage A: per (batch,level) masked sigmoid top-K + decode quads ----
__global__ void fcos_topk_level(const float* __restrict__ loc,
                                const float* __restrict__ cls,
                                const float* __restrict__ reg,
                                const int* __restrict__ imsz,
                                float* __restrict__ polyU,
                                float* __restrict__ scoreU,
                                float* __restrict__ validU,
                                int N, int K, int off) {
  __shared__ unsigned long long red[256];
  __shared__ int selList[256];
  int b = blockIdx.x, tid = threadIdx.x;
  const float* clsB = cls + (size_t)b * N;
  unsigned long long prev = ~0ull;
  for (int j = 0; j < K; ++j) {
    unsigned long long local = 0ull;
    for (int i = tid; i < N; i += 256) {
      float s = 1.0f / (1.0f + expf(-clsB[i]));
      float m = (s > PRE_T) ? s : 0.0f;
      unsigned long long key = packKey(m, (unsigned)i);
      if (key < prev && key > local) local = key;
    }
    __syncthreads();
    red[tid] = local;
    __syncthreads();
    for (int s = 128; s > 0; s >>= 1) {
      if (tid < s && red[tid + s] > red[tid]) red[tid] = red[tid + s];
      __syncthreads();
    }
    unsigned long long win = red[0];
    if (tid == 0) selList[j] = (int)keyIdx(win);
    prev = win;
  }
  __syncthreads();
  float hh = (float)imsz[b*2+0];
  float ww = (float)imsz[b*2+1];
  for (int j = tid; j < K; j += 256) {
    int idx = selList[j];
    float s = 1.0f / (1.0f + expf(-clsB[idx]));
    float val = (s > PRE_T) ? s : 0.0f;
    bool valid = (val > PRE_T);       // MIN_SIZE==0 -> bw/bh checks always pass
    float x = loc[idx*2+0], y = loc[idx*2+1];
    int g = b * T_TOT + off + j;
#pragma unroll
    for (int c = 0; c < 8; ++c) {
      float r = reg[((size_t)(b*8 + c)) * N + idx];
      float p = (((c & 1) == 0) ? x : y) - r;
      float hi = ((c & 1) == 0) ? (ww - 1.0f) : (hh - 1.0f);
      p = fminf(fmaxf(p, 0.0f), hi);
      polyU[(size_t)g*8 + c] = p;
    }
    scoreU[g] = valid ? sqrtf(val) : 0.0f;
    validU[g] = valid ? 1.0f : 0.0f;
  }
}

// ---- Stage B: per-batch stable descending sort (bitonic over 1024 keys in LDS) ----
__global__ void fcos_sort(const float* __restrict__ polyU,
                          const float* __restrict__ scoreU,
                          const float* __restrict__ validU,
                          float* __restrict__ polyS, float* __restrict__ scoreS,
                          float* __restrict__ validS, float* __restrict__ areasS) {
  __shared__ unsigned long long keys[SORT_N];
  int b = blockIdx.x, tid = threadIdx.x;
  for (int i = tid; i < SORT_N; i += 256)
    keys[i] = (i < T_TOT) ? packKey(scoreU[b*T_TOT + i], (unsigned)i) : 0ull;
  for (unsigned size = 2; size <= SORT_N; size <<= 1)
    for (unsigned stride = size >> 1; stride > 0; stride >>= 1) {
      __syncthreads();
      for (unsigned t = tid; t < SORT_N/2; t += 256) {
        unsigned lo = 2*t - (t & (stride - 1));
        unsigned hi = lo + stride;
        bool desc = ((lo & size) == 0);
        unsigned long long a = keys[lo], c = keys[hi];
        bool sw = desc ? (a < c) : (a > c);
        if (sw) { keys[lo] = c; keys[hi] = a; }
      }
    }
  __syncthreads();
  for (int i = tid; i < T_TOT; i += 256) {
    int o = (int)keyIdx(keys[i]);
    int src = b*T_TOT + o, dst = b*T_TOT + i;
    float qx[4], qy[4];
#pragma unroll
    for (int c = 0; c < 4; ++c) {
      float x = polyU[(size_t)src*8 + 2*c];
      float y = polyU[(size_t)src*8 + 2*c + 1];
      qx[c] = x; qy[c] = y;
      polyS[(size_t)dst*8 + 2*c]     = x;
      polyS[(size_t)dst*8 + 2*c + 1] = y;
    }
    scoreS[dst] = scoreU[src];
    validS[dst] = validU[src];
    areasS[dst] = quadAreaF(qx, qy);
  }
}

// ---- Stage C: pairwise quad IoU, one wave per 16x16 tile.
// union = areas_i + areas_j - inter via V_WMMA_F32_16X16X4_F32:
//   A = [areas_i | 1 | 0 | 0] (16x4), B = [1 ; areas_j ; 0 ; 0] (4x16), C = -inter.
__global__ void fcos_iou_tiles(const float* __restrict__ polyS,
                               const float* __restrict__ areasS,
                               float* __restrict__ iouMat, int tiles) {
  int lane = threadIdx.x;
  int blk = blockIdx.x;
  int b   = blk / (tiles * tiles);
  int rem = blk % (tiles * tiles);
  int ti = rem / tiles, tj = rem % tiles;

  int nn = lane & 15;
  int j = tj * 16 + nn;
  float Bx[4], By[4];
#pragma unroll
  for (int c = 0; c < 4; ++c) {
    bool ok = (j < T_TOT);
    size_t base = ((size_t)(b*T_TOT + (ok ? j : 0))) * 8;
    Bx[c] = ok ? polyS[base + 2*c]     : 0.f;
    By[c] = ok ? polyS[base + 2*c + 1] : 0.f;
  }

  float inter[8];
  v8f cfrag;
#pragma unroll
  for (int v = 0; v < 8; ++v) {
    int m = (lane < 16) ? v : (v + 8);
    int i = ti * 16 + m;
    float Axq[4], Ayq[4];
#pragma unroll
    for (int c = 0; c < 4; ++c) {
      bool ok = (i < T_TOT);
      size_t base = ((size_t)(b*T_TOT + (ok ? i : 0))) * 8;
      Axq[c] = ok ? polyS[base + 2*c]     : 0.f;
      Ayq[c] = ok ? polyS[base + 2*c + 1] : 0.f;
    }
    float iv = polyInterArea(Axq, Ayq, Bx, By);   // clip row-quad A by col-quad B
    inter[v] = iv;
    cfrag[v] = -iv;
  }

  int lm = lane & 15;
  int gi = ti * 16 + lm, gj = tj * 16 + lm;
  float aI = (lane < 16 && gi < T_TOT) ? areasS[b*T_TOT + gi] : 0.f;
  float aJ = (lane < 16 && gj < T_TOT) ? areasS[b*T_TOT + gj] : 0.f;
  v2f afrag, bfrag;
  afrag[0] = aI;                       // A: K=0 col = areas_i (lanes 0-15), K=2/3 zero
  afrag[1] = (lane < 16) ? 1.0f : 0.0f; // A: K=1 col = ones
  bfrag[0] = (lane < 16) ? 1.0f : 0.0f; // B: row K=0 = ones, rows K=2/3 zero
  bfrag[1] = aJ;                        // B: row K=1 = areas_j
  v8f uni = __builtin_amdgcn_wmma_f32_16x16x4_f32(
      false, afrag, false, bfrag, (short)0, cfrag, false, false);

#pragma unroll
  for (int v = 0; v < 8; ++v) {
    int m = (lane < 16) ? v : (v + 8);
    int i = ti * 16 + m;
    if (i < T_TOT && j < T_TOT) {
      float u = fmaxf(uni[v], 1e-9f);
      iouMat[(size_t)b*T_TOT*T_TOT + (size_t)i*T_TOT + j] = inter[v] / u;
    }
  }
}

// ---- Stage D: greedy suppression + final top-100, one workgroup per batch ----
__global__ void fcos_nms_final(const float* __restrict__ polyS,
                               const float* __restrict__ scoreS,
                               const float* __restrict__ validS,
                               const float* __restrict__ iouMat,
                               float* __restrict__ out) {
  __shared__ int keep[T_TOT];
  __shared__ unsigned long long red[256];
  int b = blockIdx.x, tid = threadIdx.x;
  const float* sc  = scoreS + b*T_TOT;
  const float* iou = iouMat + (size_t)b*T_TOT*T_TOT;
  for (int i = tid; i < T_TOT; i += 256) keep[i] = (validS[b*T_TOT + i] > 0.5f) ? 1 : 0;
  __syncthreads();
  for (int i = 0; i < T_TOT; ++i) {
    if (keep[i]) {
      for (int j2 = i + 1 + tid; j2 < T_TOT; j2 += 256)
        if (iou[(size_t)i*T_TOT + j2] > NMS_T) keep[j2] = 0;
    }
    __syncthreads();
  }
  unsigned long long prev = ~0ull;
  for (int r = 0; r < POST_N; ++r) {
    unsigned long long local = 0ull;
    for (int i = tid; i < T_TOT; i += 256) {
      float f = keep[i] ? sc[i] : -1.0f;
      unsigned long long key = packKey(f, (unsigned)i);
      if (key < prev && key > local) local = key;
    }
    __syncthreads();
    red[tid] = local;
    __syncthreads();
    for (int s = 128; s > 0; s >>= 1) {
      if (tid < s && red[tid + s] > red[tid]) red[tid] = red[tid + s];
      __syncthreads();
    }
    unsigned long long win = red[0];
    if (tid == 0) {
      int t = (int)keyIdx(win);
      float fv = keep[t] ? sc[t] : -1.0f;
      bool ok = (fv > 0.0f);
      int o = b*POST_N + r;
#pragma unroll
      for (int c = 0; c < 8; ++c)
        out[(size_t)o*8 + c] = polyS[((size_t)(b*T_TOT + t))*8 + c];
      out[NB*POST_N*8  + o] = ok ? fv : 0.0f;  // scores
      out[NB*POST_N*9  + o] = 1.0f;            // labels (C==1 -> always 1)
      out[NB*POST_N*10 + o] = ok ? 1.0f : 0.0f; // valid
    }
    prev = win;
  }
}

extern "C" void kernel_launch(void* const* d_in, const int* in_sizes, int n_in,
                              void* d_out, int out_size, void* d_ws, size_t ws_size,
                              hipStream_t stream) {
  (void)in_sizes; (void)n_in; (void)out_size; (void)ws_size;
  static const int N[5]   = {12800, 3200, 800, 208, 56};
  static const int K[5]   = {200, 200, 200, 200, 56};
  static const int OFF[5] = {0, 200, 400, 600, 800};

  float* p      = (float*)d_ws;
  float* polyU  = p; p += NB*T_TOT*8;
  float* scoreU = p; p += NB*T_TOT;
  float* validU = p; p += NB*T_TOT;
  float* polyS  = p; p += NB*T_TOT*8;
  float* scoreS = p; p += NB*T_TOT;
  float* validS = p; p += NB*T_TOT;
  float* areasS = p; p += NB*T_TOT;
  float* iouMat = p;               // NB * T_TOT * T_TOT floats (~5.9 MB)

  const int* imsz = (const int*)d_in[20];
  for (int lv = 0; lv < 5; ++lv) {
    // setup_inputs dict order: per level [locations, box_cls, box_reg, centerness]
    fcos_topk_level<<<NB, 256, 0, stream>>>(
        (const float*)d_in[lv*4 + 0], (const float*)d_in[lv*4 + 1],
        (const float*)d_in[lv*4 + 2], imsz,
        polyU, scoreU, validU, N[lv], K[lv], OFF[lv]);
  }
  fcos_sort<<<NB, 256, 0, stream>>>(polyU, scoreU, validU,
                                    polyS, scoreS, validS, areasS);
  int tiles = (T_TOT + 15) / 16;   // 54
  fcos_iou_tiles<<<NB*tiles*tiles, 32, 0, stream>>>(polyS, areasS, iouMat, tiles);
  fcos_nms_final<<<NB, 256, 0, stream>>>(polyS, scoreS, validS, iouMat, (float*)d_out);
}